// NodeNetwork_6365141533086
// MI455X (gfx1250) — compile-verified
//
#include <hip/hip_runtime.h>

typedef float v2f __attribute__((ext_vector_type(2)));
typedef float v8f __attribute__((ext_vector_type(8)));

#define NN 50000
#define FEAT 64
#define HID 16

// ---------------- degree / norm ----------------

__global__ void zero_cnt_kernel(int* __restrict__ cnt, int n) {
    int i = blockIdx.x * blockDim.x + threadIdx.x;
    if (i < n) cnt[i] = 0;
}

__global__ void count_deg_kernel(const long long* __restrict__ col, int* __restrict__ cnt, int E) {
    int e = blockIdx.x * blockDim.x + threadIdx.x;
    if (e < E) atomicAdd(&cnt[(int)col[e]], 1);
}

// In-place: reinterpret int count -> float dinv. deg = 8*cnt+1 for i<N else 1.
__global__ void make_dinv_kernel(float* __restrict__ dinv, int n) {
    int i = blockIdx.x * blockDim.x + threadIdx.x;
    if (i >= n) return;
    int c = (i < NN) ? ((const int*)dinv)[i] : 0;
    float deg = (float)(8 * c + 1);
    dinv[i] = rsqrtf(deg);
}

// ---------------- WMMA GEMMs (self-loop agg-init fused into epilogue) ----------------

// h[total x 16] = x[total x 64] @ W[64 x 16]; agg[r][f] = h[r][f]*dinv[r]^2.
// One wave per 16-row tile.
__global__ void __launch_bounds__(256) gemm_k64_kernel(const float* __restrict__ x,
                                                       const float* __restrict__ W,
                                                       const float* __restrict__ dinv,
                                                       float* __restrict__ h,
                                                       float* __restrict__ agg, int ntiles) {
    const int lane = threadIdx.x & 31;
    const int tile = blockIdx.x * 8 + (threadIdx.x >> 5);
    if (tile >= ntiles) return;            // never true (exact grid); keeps EXEC all-1s
    const int m  = lane & 15;              // row (A) / col (B,D) index
    const int kh = lane >> 4;              // which K-pair this half-wave holds
    const float* xrow = x + ((long)tile * 16 + m) * FEAT;
    v8f c = {};
#pragma unroll
    for (int kk = 0; kk < 16; ++kk) {
        const int k0 = kk * 4 + 2 * kh;
        v2f a, b;
        a.x = xrow[k0];
        a.y = xrow[k0 + 1];
        b.x = W[k0 * 16 + m];
        b.y = W[(k0 + 1) * 16 + m];
        c = __builtin_amdgcn_wmma_f32_16x16x4_f32(false, a, false, b, (short)0, c, false, false);
    }
    const float* dp = dinv + (long)tile * 16 + 8 * kh;   // this half-wave's 8 output rows
#pragma unroll
    for (int v = 0; v < 8; ++v) {
        const long r = (long)tile * 16 + v + 8 * kh;
        const float d = dp[v];
        h[r * 16 + m]   = c[v];
        agg[r * 16 + m] = c[v] * (d * d);
    }
}

// g[total x 16] = relu(aggin[total x 16] + bias) @ W[16 x 16]; relu+bias fused into A load.
// Also writes agg2-init IN PLACE over aggin (same row block per wave -> safe).
__global__ void __launch_bounds__(256) gemm_k16_relu_kernel(float* __restrict__ aggin,
                                                            const float* __restrict__ bias,
                                                            const float* __restrict__ W,
                                                            const float* __restrict__ dinv,
                                                            float* __restrict__ g, int ntiles) {
    const int lane = threadIdx.x & 31;
    const int tile = blockIdx.x * 8 + (threadIdx.x >> 5);
    if (tile >= ntiles) return;
    const int m  = lane & 15;
    const int kh = lane >> 4;
    const float* arow = aggin + ((long)tile * 16 + m) * HID;
    v8f c = {};
#pragma unroll
    for (int kk = 0; kk < 4; ++kk) {
        const int k0 = kk * 4 + 2 * kh;
        v2f a, b;
        a.x = fmaxf(arow[k0]     + bias[k0],     0.0f);
        a.y = fmaxf(arow[k0 + 1] + bias[k0 + 1], 0.0f);
        b.x = W[k0 * 16 + m];
        b.y = W[(k0 + 1) * 16 + m];
        c = __builtin_amdgcn_wmma_f32_16x16x4_f32(false, a, false, b, (short)0, c, false, false);
    }
    const float* dp = dinv + (long)tile * 16 + 8 * kh;
#pragma unroll
    for (int v = 0; v < 8; ++v) {
        const long r = (long)tile * 16 + v + 8 * kh;
        const float d = dp[v];
        g[r * 16 + m]     = c[v];
        aggin[r * 16 + m] = c[v] * (d * d);   // in-place agg2 init
    }
}

// ---------------- edge aggregation ----------------

// agg[col][f] += 8 * dinv[row]*dinv[col] * h[row][f]; 16 lanes per edge.
__global__ void edge_scatter_kernel(const long long* __restrict__ row,
                                    const long long* __restrict__ col,
                                    const float* __restrict__ dinv,
                                    const float* __restrict__ h,
                                    float* __restrict__ agg, int E) {
    int t = blockIdx.x * blockDim.x + threadIdx.x;
    if (t >= E * 16) return;
    int e = t >> 4, f = t & 15;
    int r = (int)row[e];
    int c = (int)col[e];
    float w = 8.0f * dinv[r] * dinv[c];
    atomicAdd(&agg[(long)c * 16 + f], h[(long)r * 16 + f] * w);
}

// ---------------- final FC (relu + b2 fused) ----------------

__global__ void fc_kernel(const float* __restrict__ agg2, const float* __restrict__ b2,
                          const float* __restrict__ Wfc, const float* __restrict__ bfc,
                          float* __restrict__ out, int n) {
    int i = blockIdx.x * blockDim.x + threadIdx.x;
    if (i >= n) return;
    const float* r = agg2 + (long)i * 16;
    float acc = bfc[0];
#pragma unroll
    for (int f = 0; f < 16; ++f)
        acc += fmaxf(r[f] + b2[f], 0.0f) * Wfc[f];
    out[i] = acc;
}

// ---------------- launch ----------------

extern "C" void kernel_launch(void* const* d_in, const int* in_sizes, int n_in,
                              void* d_out, int out_size, void* d_ws, size_t ws_size,
                              hipStream_t stream) {
    const float*     x   = (const float*)d_in[0];
    const long long* ei  = (const long long*)d_in[1];
    const float*     W1  = (const float*)d_in[2];
    const float*     b1  = (const float*)d_in[3];
    const float*     W2  = (const float*)d_in[4];
    const float*     b2  = (const float*)d_in[5];
    const float*     Wfc = (const float*)d_in[6];
    const float*     bfc = (const float*)d_in[7];
    float* out = (float*)d_out;

    const int E     = in_sizes[1] / 2;        // 800000
    const int total = in_sizes[0] / FEAT;     // 400000
    const long long* rowi = ei;
    const long long* coli = ei + E;

    float* bufA = (float*)d_ws;                       // total*16 floats: h, then g
    float* bufB = bufA + (size_t)total * HID;         // total*16 floats: agg1, then agg2
    float* dinv = bufB + (size_t)total * HID;         // total floats (int cnt first)
    int*   cnt  = (int*)dinv;

    const int ntiles = total / 16;                    // 25000
    const int gblk   = (ntiles + 7) / 8;              // 3125 blocks x 8 waves

    // degrees -> dinv
    zero_cnt_kernel<<<(total + 255) / 256, 256, 0, stream>>>(cnt, total);
    count_deg_kernel<<<(E + 255) / 256, 256, 0, stream>>>(coli, cnt, E);
    make_dinv_kernel<<<(total + 255) / 256, 256, 0, stream>>>(dinv, total);

    // layer 1: h = x @ W1 -> bufA, agg1 self-loop init -> bufB; then edge scatter into bufB
    gemm_k64_kernel<<<gblk, 256, 0, stream>>>(x, W1, dinv, bufA, bufB, ntiles);
    edge_scatter_kernel<<<(E * 16 + 255) / 256, 256, 0, stream>>>(rowi, coli, dinv, bufA, bufB, E);

    // layer 2: g = relu(agg1 + b1) @ W2 -> bufA, agg2 init in place in bufB; scatter into bufB
    gemm_k16_relu_kernel<<<gblk, 256, 0, stream>>>(bufB, b1, W2, dinv, bufA, ntiles);
    edge_scatter_kernel<<<(E * 16 + 255) / 256, 256, 0, stream>>>(rowi, coli, dinv, bufA, bufB, E);

    // head: out = relu(agg2 + b2) @ Wfc + bfc
    fc_kernel<<<(total + 255) / 256, 256, 0, stream>>>(bufB, b2, Wfc, bfc, out, total);
}